// QuantumConvLayer_57123065037201
// MI455X (gfx1250) — compile-verified
//
#include <hip/hip_runtime.h>

// ---------------------------------------------------------------------------
// QuantumConvLayer for MI455X (gfx1250, wave32).
//   image [64,1,512,512] f32  ->  out [64,4,256,256] f32
// One wave = one output row (b, y): 16 tiles x 16 patches = 256 patches.
//   psi^T[state, patch] = U[16x16] x amp^T[state_in, patch]
// as 4 chained V_WMMA_F32_16X16X4_F32 per tile (fp32, K=16 in steps of 4).
// PauliZ reduction: packed (V_PK_*_F32) butterfly + 2 ds_swizzle exchanges.
// ---------------------------------------------------------------------------

typedef __attribute__((ext_vector_type(2))) float v2f;
typedef __attribute__((ext_vector_type(8))) float v8f;

#define OW 256
#define OH 256
#define WAVES_PER_BLOCK 8
#define TILES_PER_WAVE 16   // == OW/16 : one full output row per wave
#define NBLOCKS 2048        // 2048*8 waves = 16384 = 64*256 rows

// ---------------------------------------------------------------------------
// Prologue: build the 16x16 layer unitary U from rand_params (4 RY + CNOT ring).
// One wave, 16 lanes; lane n owns column n (all gate ops are column-local).
// Stores U row-major to ws[r*16 + n].
// ---------------------------------------------------------------------------
__global__ __launch_bounds__(16) void build_unitary(const float* __restrict__ rp,
                                                    float* __restrict__ Uws) {
  const int n = threadIdx.x;
  float col[16];
#pragma unroll
  for (int r = 0; r < 16; ++r) col[r] = (r == n) ? 1.0f : 0.0f;

  // RY(theta_i) on wire i (wire 0 = MSB).  v_cos/v_sin take revolutions.
#pragma unroll
  for (int i = 0; i < 4; ++i) {
    const float half = 0.5f * rp[i];
    const float c = __builtin_amdgcn_cosf(half * 0.15915494309189535f);
    const float s = __builtin_amdgcn_sinf(half * 0.15915494309189535f);
    const int mask = 1 << (3 - i);
#pragma unroll
    for (int r = 0; r < 16; ++r) {
      if (r & mask) continue;
      const int r1 = r | mask;
      const float a = col[r], b = col[r1];
      col[r]  = c * a - s * b;   // G = [[c,-s],[s,c]]
      col[r1] = s * a + c * b;
    }
  }
  // CNOT ring (0,1),(1,2),(2,3),(3,0): row permutation within each column.
  const int ctrl[4] = {0, 1, 2, 3};
  const int tgt[4]  = {1, 2, 3, 0};
#pragma unroll
  for (int g = 0; g < 4; ++g) {
    const int cm = 1 << (3 - ctrl[g]);
    const int tm = 1 << (3 - tgt[g]);
    float tmp[16];
#pragma unroll
    for (int r = 0; r < 16; ++r) tmp[r] = col[(r & cm) ? (r ^ tm) : r];
#pragma unroll
    for (int r = 0; r < 16; ++r) col[r] = tmp[r];
  }
#pragma unroll
  for (int r = 0; r < 16; ++r) Uws[r * 16 + n] = col[r];
}

// ---------------------------------------------------------------------------
// Main kernel.
// ---------------------------------------------------------------------------
__global__ __launch_bounds__(256) void qconv_wmma(const float* __restrict__ img,
                                                  const float* __restrict__ Uws,
                                                  float* __restrict__ out) {
  const int lane = threadIdx.x & 31;
  const int wave = blockIdx.x * WAVES_PER_BLOCK + (threadIdx.x >> 5);
  const int m  = lane & 15;   // patch index within tile (WMMA N); also A-row M
  const bool hi = (lane >> 4) != 0;  // K parity half: lo {0,1}, hi {2,3}

  // A = U fragments (loop invariant). A[M=m, K=4t+2*hi+{0,1}] = U[m, k].
  v2f a0, a1, a2, a3;
  {
    const float* Urow = Uws + m * 16 + (hi ? 2 : 0);
    a0 = *(const v2f*)(Urow + 0);
    a1 = *(const v2f*)(Urow + 4);
    a2 = *(const v2f*)(Urow + 8);
    a3 = *(const v2f*)(Urow + 12);
  }

  // This wave's output row: b = wave>>8, y = wave&255 (loop invariant).
  const int b = wave >> 8;
  const int y = wave & (OH - 1);

  // Per-lane base pointers; every tile access is base + compile-time offset.
  const float* pin = img + ((size_t)(b * 512 + 2 * y) * 512 + 2 * m);
  const size_t plane = (size_t)OH * OW;                       // 65536 floats
  float* outp = out + ((((size_t)b * 4 + (hi ? 2 : 0)) * OH + y) * OW + m);

  const float hoff = hi ? 0.0f : 0.25f;    // cos(x) = sin(x + 1/4 rev)
  const float sgn0 = hi ? -1.0f : 1.0f;    // wire0 sign = bit3 of state = hi

#pragma unroll 4
  for (int ti = 0; ti < TILES_PER_WAVE; ++ti) {
    // 2x2 patch: rows 2y,2y+1, cols 2*(ti*16+m)+{0,1}  -> immediate offsets.
    const v2f r0 = *(const v2f*)(pin + ti * 32);
    const v2f r1 = *(const v2f*)(pin + ti * 32 + 512);

    // RY(pi*phi): per-wire (cos, sin) of (pi/2)*phi; v_sin takes revolutions,
    // so the argument is phi/4.  Packed scale (v_pk_mul_f32), 7 trans total.
    const v2f h01 = r0 * 0.25f;
    const v2f h23 = r1 * 0.25f;
    const float c0 = __builtin_amdgcn_cosf(h01.x), s0 = __builtin_amdgcn_sinf(h01.x);
    const float c1 = __builtin_amdgcn_cosf(h01.y), s1 = __builtin_amdgcn_sinf(h01.y);
    const float t2 = __builtin_amdgcn_sinf(h23.x + hoff);  // lo: cos2, hi: sin2
    const float c3 = __builtin_amdgcn_cosf(h23.y), s3 = __builtin_amdgcn_sinf(h23.y);

    // amp[k] = q_a(w0) q_b(w1) q_c(w2) q_d(w3), k = a*8+b*4+c*2+d.
    // B chunk t holds k = 4t + 2*hi + {0,1}  ->  c == hi, d == parity.
    const v2f cs1 = {c1, s1};
    const v2f cs3 = {c3, s3};
    const v2f u    = cs3 * t2;          // (t2*c3, t2*s3)
    const v2f pabA = cs1 * c0;          // (c0c1, c0s1)
    const v2f pabB = cs1 * s0;          // (s0c1, s0s1)
    const v2f b0 = u * pabA.x;
    const v2f b1 = u * pabA.y;
    const v2f b2 = u * pabB.x;
    const v2f b3 = u * pabB.y;

    // psi^T = U x amp^T : 4 chained fp32 WMMAs.
    v8f acc = {};
    acc = __builtin_amdgcn_wmma_f32_16x16x4_f32(false, a0, false, b0, (short)0, acc, false, false);
    acc = __builtin_amdgcn_wmma_f32_16x16x4_f32(false, a1, false, b1, (short)0, acc, false, false);
    acc = __builtin_amdgcn_wmma_f32_16x16x4_f32(false, a2, false, b2, (short)0, acc, false, false);
    acc = __builtin_amdgcn_wmma_f32_16x16x4_f32(false, a3, false, b3, (short)0, acc, false, false);

    // Lane holds psi[state = r + 8*hi, patch = m] in acc[r].
    // probs = psi^2 as 4 packed muls on (even,odd) pairs.
    v2f P0 = {acc[0], acc[1]};  P0 *= P0;   // (p0,p1)
    v2f P1 = {acc[2], acc[3]};  P1 *= P1;   // (p2,p3)
    v2f P2 = {acc[4], acc[5]};  P2 *= P2;   // (p4,p5)
    v2f P3 = {acc[6], acc[7]};  P3 *= P3;   // (p6,p7)

    // Packed PauliZ butterfly over this lane's 8 states (sign = state bit).
    const v2f B0 = P0 + P2;                 // (p0+p4, p1+p5)
    const v2f B1 = P1 + P3;                 // (p2+p6, p3+p7)
    const v2f C0 = B0 + B1;                 // (evens, odds)
    const v2f C1 = B0 - B1;
    const v2f E  = (P0 - P2) + (P1 - P3);
    const float w0p = sgn0 * (C0.x + C0.y); // wire0 partial: bit3 = hi
    const float w1p = E.x + E.y;            // wire1 partial: bit2
    const float w2p = C1.x + C1.y;          // wire2 partial: bit1
    const float w3p = C0.x - C0.y;          // wire3 partial: bit0

    // Each lane stores only 2 wires (lo: 0,1 / hi: 2,3), so exchange exactly
    // what the partner lane (l^16) needs: 2 ds_swizzle SWAPX16 instead of 4.
    //   send X  = hi ? w0p : w2p   keep Y  = hi ? w2p : w0p
    //   send X2 = hi ? w1p : w3p   keep Y2 = hi ? w3p : w1p
    const float X  = hi ? w0p : w2p;
    const float Y  = hi ? w2p : w0p;
    const float X2 = hi ? w1p : w3p;
    const float Y2 = hi ? w3p : w1p;
    const float st0 = Y  + __int_as_float(__builtin_amdgcn_ds_swizzle(__float_as_int(X),  0x401F));
    const float st1 = Y2 + __int_as_float(__builtin_amdgcn_ds_swizzle(__float_as_int(X2), 0x401F));

    // lo lanes: (o0, o1); hi lanes: (o2, o3).  Immediate-offset stores.
    outp[ti * 16]         = st0;
    outp[ti * 16 + plane] = st1;
  }
}

// ---------------------------------------------------------------------------
extern "C" void kernel_launch(void* const* d_in, const int* in_sizes, int n_in,
                              void* d_out, int out_size, void* d_ws, size_t ws_size,
                              hipStream_t stream) {
  const float* image = (const float*)d_in[0];   // [64,1,512,512] f32
  const float* rp    = (const float*)d_in[1];   // [1,4] f32
  float* out = (float*)d_out;                   // [64,4,256,256] f32
  float* Uws = (float*)d_ws;                    // 256 floats of scratch

  build_unitary<<<1, 16, 0, stream>>>(rp, Uws);
  qconv_wmma<<<NBLOCKS, 256, 0, stream>>>(image, Uws, out);
}